// SELayer_drop_68891275428392
// MI455X (gfx1250) — compile-verified
//
#include <hip/hip_runtime.h>

// MI455X / gfx1250: wave32, WMMA (not MFMA).
// Roofline: dominated by 2 passes over x (103 MB) -> ~206-308 MB HBM traffic
// -> ~9-13 us at 23.3 TB/s. MLP is ~4 MFLOP: done with fp32 WMMA
// (V_WMMA_F32_16X16X4_F32) for exact fp32 fidelity on the matrix pipe.

typedef __attribute__((ext_vector_type(2))) float v2f;
typedef __attribute__((ext_vector_type(8))) float v8f;
typedef __attribute__((ext_vector_type(4))) float f4;

#define B_   64
#define C_   512
#define R_   32
#define HW_  784          // 28*28
#define HW4_ 196          // 784/4
#define K_   256          // int(C * 0.5) kept channels

// ---------------------------------------------------------------------------
// Kernel 1: global average pool. One wave32 per (b,c) channel.
// 8 waves per 256-thread block -> 4096 blocks cover 64*512 channels.
// ---------------------------------------------------------------------------
__global__ __launch_bounds__(256) void se_gap_kernel(const float* __restrict__ x,
                                                     float* __restrict__ y) {
  const int lane = threadIdx.x & 31;
  const int wave = threadIdx.x >> 5;
  const int chan = blockIdx.x * 8 + wave;          // 0 .. 32767
  const f4* xb = (const f4*)(x + (size_t)chan * HW_);
  float s = 0.0f;
  #pragma unroll
  for (int i = lane; i < HW4_; i += 32) {          // 196 float4 per channel
    f4 v = xb[i];
    s += (v.x + v.y) + (v.z + v.w);
  }
  // wave32 butterfly reduction
  for (int off = 16; off > 0; off >>= 1) s += __shfl_down(s, off, 32);
  if (lane == 0) y[chan] = s * (1.0f / (float)HW_);
}

// ---------------------------------------------------------------------------
// Kernel 2: SE MLP with fp32 WMMA.
//   T = relu(Y(64x512) @ W1^T(512x32))     -> kept in LDS
//   G = sigmoid(T(64x32) @ W2^T(32x512))   -> gates in workspace
// Fragment layouts per CDNA5 ISA 7.12.2:
//   A (16x4 f32, 2 VGPRs): lane L -> row M=L%16, K = 2*(L/16)+v
//   B (4x16 f32, 2 VGPRs): lane L -> col N=L%16, K = 2*(L/16)+v
//   C (16x16 f32, 8 VGPRs): lane L -> M = v + 8*(L/16), N = L%16
// ---------------------------------------------------------------------------
__global__ __launch_bounds__(256) void se_mlp_kernel(const float* __restrict__ y,   // 64x512
                                                     const float* __restrict__ w1,  // 32x512 (row-major [out,in])
                                                     const float* __restrict__ w2,  // 512x32
                                                     float* __restrict__ gate) {    // 64x512
  __shared__ float T[B_ * R_];                     // 8 KB
  const int lane = threadIdx.x & 31;
  const int wave = threadIdx.x >> 5;               // 0..7
  const int lm   = lane & 15;
  const int lhi  = lane >> 4;                      // 0 or 1
  const int kk   = lhi << 1;                       // 0 or 2

  // ---- GEMM1: 4 M-tiles x 2 N-tiles = 8 tiles, one per wave ----
  {
    const int m0 = (wave >> 1) << 4;
    const int n0 = (wave & 1) << 4;
    v8f acc = {};
    const float* arow = y  + (size_t)(m0 + lm) * C_;   // A row (Y)
    const float* brow = w1 + (size_t)(n0 + lm) * C_;   // B col: W1^T[k][n] = w1[n][k]
    for (int k0 = 0; k0 < C_; k0 += 4) {
      v2f a, b;
      a[0] = arow[k0 + kk + 0];
      a[1] = arow[k0 + kk + 1];
      b[0] = brow[k0 + kk + 0];
      b[1] = brow[k0 + kk + 1];
      acc = __builtin_amdgcn_wmma_f32_16x16x4_f32(
          /*neg_a=*/false, a, /*neg_b=*/false, b,
          /*c_mod=*/(short)0, acc, /*reuse_a=*/false, /*reuse_b=*/false);
    }
    #pragma unroll
    for (int v = 0; v < 8; ++v) {
      const int M = m0 + v + (lhi << 3);
      const int N = n0 + lm;
      T[M * R_ + N] = fmaxf(acc[v], 0.0f);             // ReLU
    }
  }
  __syncthreads();

  // ---- GEMM2: 4 M-tiles x 32 N-tiles = 128 tiles, 16 per wave ----
  for (int t = wave; t < 128; t += 8) {                // uniform per wave -> EXEC all 1s
    const int m0 = (t >> 5) << 4;
    const int n0 = (t & 31) << 4;
    v8f acc = {};
    const float* arow = &T[(m0 + lm) * R_];
    const float* brow = w2 + (size_t)(n0 + lm) * R_;   // W2^T[r][c] = w2[c][r]
    #pragma unroll
    for (int k0 = 0; k0 < R_; k0 += 4) {
      v2f a, b;
      a[0] = arow[k0 + kk + 0];
      a[1] = arow[k0 + kk + 1];
      b[0] = brow[k0 + kk + 0];
      b[1] = brow[k0 + kk + 1];
      acc = __builtin_amdgcn_wmma_f32_16x16x4_f32(
          false, a, false, b, (short)0, acc, false, false);
    }
    #pragma unroll
    for (int v = 0; v < 8; ++v) {
      const int M = m0 + v + (lhi << 3);
      const int N = n0 + lm;
      gate[M * C_ + N] = 1.0f / (1.0f + __expf(-acc[v]));   // sigmoid
    }
  }
}

// ---------------------------------------------------------------------------
// Kernel 3: 'larger' mask, top-256-of-512 per row, via rank counting.
// Tie-break matches top_k: equal values keep the lower index first.
// One block per batch row.
// ---------------------------------------------------------------------------
__global__ __launch_bounds__(256) void se_mask_kernel(float* __restrict__ gate) {
  __shared__ float row[C_];
  float* g = gate + (size_t)blockIdx.x * C_;
  for (int i = threadIdx.x; i < C_; i += blockDim.x) row[i] = g[i];
  __syncthreads();
  for (int i = threadIdx.x; i < C_; i += blockDim.x) {
    const float v = row[i];
    int rank = 0;
    for (int j = 0; j < C_; ++j) {
      const float u = row[j];
      rank += (u > v) || ((u == v) && (j < i));
    }
    g[i] = (rank < K_) ? v : 0.0f;
  }
}

// ---------------------------------------------------------------------------
// Kernel 4: out = x * gate[b,c] broadcast. float4, bandwidth-bound pass;
// x should still be L2-resident (103 MB < 192 MB L2) after the GAP pass.
// ---------------------------------------------------------------------------
__global__ __launch_bounds__(256) void se_scale_kernel(const float* __restrict__ x,
                                                       const float* __restrict__ gate,
                                                       float* __restrict__ out) {
  const unsigned idx = blockIdx.x * 256u + threadIdx.x;   // over float4 elements
  const unsigned chan = idx / HW4_;                        // (b*C + c)
  const float g = gate[chan];
  const f4* x4 = (const f4*)x;
  f4* o4 = (f4*)out;
  f4 v = x4[idx];
  v.x *= g; v.y *= g; v.z *= g; v.w *= g;
  o4[idx] = v;
}

// ---------------------------------------------------------------------------
extern "C" void kernel_launch(void* const* d_in, const int* in_sizes, int n_in,
                              void* d_out, int out_size, void* d_ws, size_t ws_size,
                              hipStream_t stream) {
  const float* x  = (const float*)d_in[0];   // (64,512,28,28)
  const float* w1 = (const float*)d_in[1];   // (32,512)
  const float* w2 = (const float*)d_in[2];   // (512,32)
  float* out  = (float*)d_out;

  float* y    = (float*)d_ws;                // 64*512 floats (pooled means)
  float* gate = y + B_ * C_;                 // 64*512 floats (gates, masked in place)

  // 1) global average pool: 32768 channels / 8 waves per block
  se_gap_kernel<<<(B_ * C_) / 8, 256, 0, stream>>>(x, y);
  // 2) SE MLP (WMMA f32): single 256-thread block
  se_mlp_kernel<<<1, 256, 0, stream>>>(y, w1, w2, gate);
  // 3) top-k mask per batch row
  se_mask_kernel<<<B_, 256, 0, stream>>>(gate);
  // 4) broadcast scale: 64*512*196 float4 = 25088 blocks * 256 threads exactly
  se_scale_kernel<<<(B_ * C_ * HW4_) / 256, 256, 0, stream>>>(x, gate, out);
}